// RWKV_RNN_71201967833725
// MI455X (gfx1250) — compile-verified
//
#include <hip/hip_runtime.h>

// RWKV single-token forward, MI455X (gfx1250), persistent cooperative kernel.
// Bandwidth-bound: 1.52 GB of f32 weights streamed once (~65us @ 23.3 TB/s).
// All GEMVs run through V_WMMA_F32_16X16X4_F32 with broadcast-B trick.
// v2: K-loop unrolled x8 with dual accumulator chains for deeper MLP.

#define RL 24
#define RE 1024
#define RH 4096
#define RV 50277
#define TPB 256
#define NWG 256
#define NWAVES ((NWG * TPB) / 32)
#define P2 8    // K-split for att k/v/r GEMVs
#define P4 16   // K-split for att o GEMV
#define P5R 8   // K-split for ffn r GEMV
#define P5K 4   // K-split for ffn k GEMV
#define P6 16   // K-split for ffn v GEMV

typedef float v2f __attribute__((ext_vector_type(2)));
typedef float v8f __attribute__((ext_vector_type(8)));

struct Params {
  const float *ctx, *state, *ln0w, *ln0b, *ln1w, *ln1b, *ln2w, *ln2b;
  const float *td, *tf, *tmk, *tmv, *tmr, *kw, *vw, *rw, *ow;
  const float *ftmk, *ftmr, *fkw, *frw, *fvw, *lnoutw, *lnoutb, *head;
  float *logits, *ostate;
  float *x0, *x1, *pk, *pv, *pr, *po, *pfr, *pkk, *pfv;
  unsigned *cnt;
};

__device__ __forceinline__ float wred(float v) {
#pragma unroll
  for (int o = 16; o; o >>= 1) v += __shfl_xor(v, o, 32);
  return v;
}

// block-wide sum of (s, s2) over 8 waves; red needs >= 18 floats
__device__ __forceinline__ void bred2(float &s, float &s2, float *red) {
  s = wred(s); s2 = wred(s2);
  int w8 = threadIdx.x >> 5;
  if ((threadIdx.x & 31) == 0) { red[w8] = s; red[8 + w8] = s2; }
  __syncthreads();
  if (threadIdx.x == 0) {
    float a = 0, b = 0;
#pragma unroll
    for (int i = 0; i < 8; ++i) { a += red[i]; b += red[8 + i]; }
    red[16] = a; red[17] = b;
  }
  __syncthreads();
  s = red[16]; s2 = red[17];
}

// grid-wide barrier: counter monotonically increases; target = phase*NWG
__device__ __forceinline__ void gsync(unsigned *cnt, unsigned target) {
  __threadfence();
  __syncthreads();
  if (threadIdx.x == 0) {
    __hip_atomic_fetch_add(cnt, 1u, __ATOMIC_RELEASE, __HIP_MEMORY_SCOPE_AGENT);
    while (__hip_atomic_load(cnt, __ATOMIC_ACQUIRE, __HIP_MEMORY_SCOPE_AGENT) < target)
      __builtin_amdgcn_s_sleep(1);
  }
  __syncthreads();
  __threadfence();  // invalidate so post-barrier loads see fresh data
}

// One wave computes y[row0..row0+15] (partial K range) = W[row, k0:k0+klen] . x
// via V_WMMA_F32_16X16X4_F32. B is the x-vector broadcast across all 16 columns,
// so D[m][*] all equal y[m]; lane 0 holds rows row0..+7, lane 16 rows +8..+15.
// Two independent accumulator chains + x8 unroll -> 8 clause-batched b64 loads
// in flight per wave (2KB) to cover HBM latency. klen must be a multiple of 8.
__device__ __forceinline__ void gemv16(const float *__restrict__ W, int ldw,
                                       int nrows, const float *xv, int row0,
                                       int k0, int klen,
                                       float *__restrict__ ydst) {
  const int lane = threadIdx.x & 31;
  const int half = lane >> 4;  // A layout: lanes 0-15 K=0,1 ; lanes 16-31 K=2,3
  const int r = lane & 15;
  int row = row0 + r;
  if (row >= nrows) row = nrows - 1;  // clamp (tail tiles); stores are guarded
  const float *wp = W + (size_t)row * ldw + k0 + 2 * half;
  const float *xp = xv + k0 + 2 * half;
  v8f c0 = {}, c1 = {};
#pragma unroll 4
  for (int k = 0; k < klen; k += 8) {
    v2f a0 = *(const v2f *)(wp + k);
    v2f a1 = *(const v2f *)(wp + k + 4);
    v2f b0 = *(const v2f *)(xp + k);
    v2f b1 = *(const v2f *)(xp + k + 4);
    c0 = __builtin_amdgcn_wmma_f32_16x16x4_f32(false, a0, false, b0, (short)0,
                                               c0, false, false);
    c1 = __builtin_amdgcn_wmma_f32_16x16x4_f32(false, a1, false, b1, (short)0,
                                               c1, false, false);
  }
  v8f c = c0 + c1;
  if (r == 0) {
    int base = row0 + 8 * half;
#pragma unroll
    for (int g = 0; g < 8; ++g) {
      int rr = base + g;
      if (rr < nrows) ydst[rr] = c[g];
    }
  }
}

__global__ void __launch_bounds__(TPB) rwkv_fused(Params p) {
  __shared__ float sh[RH];    // 16KB multipurpose activation staging
  __shared__ float red[24];
  float *sh3 = sh + 3 * RE;
  const int tid = threadIdx.x;
  const int wid = (blockIdx.x * TPB + tid) >> 5;
  unsigned phase = 0;

  // ---------- stage0: x0 = LN(ctx, ln0) ----------
  if (blockIdx.x == 0) {
    float s = 0, s2 = 0;
    for (int i = tid; i < RE; i += TPB) { float v = p.ctx[i]; s += v; s2 += v * v; }
    bred2(s, s2, red);
    float mu = s * (1.f / RE), var = s2 * (1.f / RE) - mu * mu;
    float rs = rsqrtf(var + 1e-5f);
    for (int i = tid; i < RE; i += TPB)
      p.x0[i] = (p.ctx[i] - mu) * rs * p.ln0w[i] + p.ln0b[i];
  }
  ++phase; gsync(p.cnt, phase * NWG);

  for (int l = 0; l < RL; ++l) {
    const float *stl = p.state + (size_t)(l * 5) * RE;
    float *ostl = p.ostate + (size_t)(l * 5) * RE;

    // ===== S2 prologue: finalize x (l>0), LN1, time-mix into LDS =====
    {
      float s = 0, s2 = 0;
      for (int i = tid; i < RE; i += TPB) {
        float xv_;
        if (l == 0) {
          xv_ = p.x0[i];
        } else {
          float fr = 0, fv = 0;
#pragma unroll
          for (int q = 0; q < P5R; ++q) fr += p.pfr[q * RE + i];
#pragma unroll
          for (int q = 0; q < P6; ++q) fv += p.pfv[q * RE + i];
          xv_ = p.x1[i] + fv / (1.f + __expf(-fr));  // x += sig(fr)*(fvw@kk)
          if (blockIdx.x == 0) p.x0[i] = xv_;        // persist residual
        }
        sh3[i] = xv_; s += xv_; s2 += xv_ * xv_;
      }
      bred2(s, s2, red);
      float mu = s * (1.f / RE), var = s2 * (1.f / RE) - mu * mu;
      float rs = rsqrtf(var + 1e-5f);
      const float *l1w = p.ln1w + l * RE, *l1b = p.ln1b + l * RE;
      const float *tmk = p.tmk + l * RE, *tmv = p.tmv + l * RE, *tmr = p.tmr + l * RE;
      const float *st1 = stl + RE;
      for (int i = tid; i < RE; i += TPB) {
        float xn = (sh3[i] - mu) * rs * l1w[i] + l1b[i];
        float sp = st1[i];
        sh[i]          = xn * tmk[i] + sp * (1.f - tmk[i]);  // xk
        sh[RE + i]     = xn * tmv[i] + sp * (1.f - tmv[i]);  // xv
        sh[2 * RE + i] = xn * tmr[i] + sp * (1.f - tmr[i]);  // xr
        if (blockIdx.x == 0) ostl[RE + i] = xn;  // new_state row 1
      }
      __syncthreads();
    }
    // ===== S2 gemv: k, v, r_pre partials (3 x 64 tiles x P2 K-chunks) =====
    {
      const size_t wo = (size_t)l * RE * RE;
      for (int u = wid; u < 3 * 64 * P2; u += NWAVES) {
        int mat = u / (64 * P2);
        int rem = u - mat * (64 * P2);
        int tile = rem / P2, part = rem - tile * P2;
        const float *W = (mat == 0 ? p.kw : (mat == 1 ? p.vw : p.rw)) + wo;
        float *yp = (mat == 0 ? p.pk : (mat == 1 ? p.pv : p.pr)) + part * RE;
        gemv16(W, RE, RE, sh + mat * RE, tile * 16, part * (RE / P2), RE / P2, yp);
      }
    }
    ++phase; gsync(p.cnt, phase * NWG);

    // ===== S4 prologue: WKV elementwise (sum partials), state rows 2..4 =====
    {
      const float *td = p.td + l * RE, *tf = p.tf + l * RE;
      const float *aa = stl + 2 * RE, *bb = stl + 3 * RE, *pp = stl + 4 * RE;
      for (int i = tid; i < RE; i += TPB) {
        float kk = 0, vv = 0, rr = 0;
#pragma unroll
        for (int q = 0; q < P2; ++q) {
          kk += p.pk[q * RE + i]; vv += p.pv[q * RE + i]; rr += p.pr[q * RE + i];
        }
        float A = aa[i], B = bb[i], P_ = pp[i];
        float ww = tf[i] + kk;
        float pm = fmaxf(P_, ww);
        float e1 = __expf(P_ - pm), e2 = __expf(ww - pm);
        float a = e1 * A + e2 * vv, b = e1 * B + e2;
        float r = 1.f / (1.f + __expf(-rr));
        sh[i] = r * a / b;  // wkv
        float ww2 = P_ + td[i];
        float p2 = fmaxf(ww2, kk);
        float e1b = __expf(ww2 - p2), e2b = __expf(kk - p2);
        if (blockIdx.x == 0) {
          ostl[2 * RE + i] = e1b * A + e2b * vv;
          ostl[3 * RE + i] = e1b * B + e2b;
          ostl[4 * RE + i] = p2;
        }
      }
      __syncthreads();
    }
    // ===== S4 gemv: ow @ wkv partials =====
    {
      const float *W = p.ow + (size_t)l * RE * RE;
      for (int u = wid; u < 64 * P4; u += NWAVES) {
        int tile = u / P4, part = u - tile * P4;
        gemv16(W, RE, RE, sh, tile * 16, part * (RE / P4), RE / P4, p.po + part * RE);
      }
    }
    ++phase; gsync(p.cnt, phase * NWG);

    // ===== S5 prologue: x += att out; LN2; channel-mix into LDS =====
    {
      float s = 0, s2 = 0;
      for (int i = tid; i < RE; i += TPB) {
        float xm = p.x0[i];
#pragma unroll
        for (int q = 0; q < P4; ++q) xm += p.po[q * RE + i];
        sh3[i] = xm; s += xm; s2 += xm * xm;
        if (blockIdx.x == 0) p.x1[i] = xm;  // persist mid-residual
      }
      bred2(s, s2, red);
      float mu = s * (1.f / RE), var = s2 * (1.f / RE) - mu * mu;
      float rs = rsqrtf(var + 1e-5f);
      const float *l2w = p.ln2w + l * RE, *l2b = p.ln2b + l * RE;
      const float *ftmk = p.ftmk + l * RE, *ftmr = p.ftmr + l * RE;
      const float *st0 = stl;
      for (int i = tid; i < RE; i += TPB) {
        float xn2 = (sh3[i] - mu) * rs * l2w[i] + l2b[i];
        float s0v = st0[i];
        sh[i]      = xn2 * ftmk[i] + s0v * (1.f - ftmk[i]);  // fxk
        sh[RE + i] = xn2 * ftmr[i] + s0v * (1.f - ftmr[i]);  // fxr
        if (blockIdx.x == 0) ostl[i] = xn2;  // new_state row 0
      }
      __syncthreads();
    }
    // ===== S5 gemv: fr_pre (E rows) + kk_pre (H rows) partials =====
    {
      const float *Wr = p.frw + (size_t)l * RE * RE;
      const float *Wk = p.fkw + (size_t)l * RH * RE;
      const int UR = 64 * P5R;          // 512 units
      const int UK = (RH / 16) * P5K;   // 1024 units
      for (int u = wid; u < UR + UK; u += NWAVES) {
        if (u < UR) {
          int tile = u / P5R, part = u - tile * P5R;
          gemv16(Wr, RE, RE, sh + RE, tile * 16, part * (RE / P5R), RE / P5R,
                 p.pfr + part * RE);
        } else {
          int u2 = u - UR;
          int tile = u2 / P5K, part = u2 - tile * P5K;
          gemv16(Wk, RE, RH, sh, tile * 16, part * (RE / P5K), RE / P5K,
                 p.pkk + part * RH);
        }
      }
    }
    ++phase; gsync(p.cnt, phase * NWG);

    // ===== S6 prologue: kk = relu(sum partials)^2 into LDS (16KB) =====
    {
      for (int i = tid; i < RH; i += TPB) {
        float s = 0;
#pragma unroll
        for (int q = 0; q < P5K; ++q) s += p.pkk[q * RH + i];
        s = fmaxf(s, 0.f);
        sh[i] = s * s;
      }
      __syncthreads();
    }
    // ===== S6 gemv: fvw @ kk partials (E rows, K=H) =====
    {
      const float *W = p.fvw + (size_t)l * RE * RH;
      for (int u = wid; u < 64 * P6; u += NWAVES) {
        int tile = u / P6, part = u - tile * P6;
        gemv16(W, RH, RE, sh, tile * 16, part * (RH / P6), RH / P6,
               p.pfv + part * RE);
      }
    }
    ++phase; gsync(p.cnt, phase * NWG);
  }

  // ---------- head: xf = x + sig(fr)*fv; LN_out; logits = head @ xf ----------
  {
    float s = 0, s2 = 0;
    for (int i = tid; i < RE; i += TPB) {
      float fr = 0, fv = 0;
#pragma unroll
      for (int q = 0; q < P5R; ++q) fr += p.pfr[q * RE + i];
#pragma unroll
      for (int q = 0; q < P6; ++q) fv += p.pfv[q * RE + i];
      float xf = p.x1[i] + fv / (1.f + __expf(-fr));
      sh3[i] = xf; s += xf; s2 += xf * xf;
    }
    bred2(s, s2, red);
    float mu = s * (1.f / RE), var = s2 * (1.f / RE) - mu * mu;
    float rs = rsqrtf(var + 1e-5f);
    for (int i = tid; i < RE; i += TPB)
      sh[i] = (sh3[i] - mu) * rs * p.lnoutw[i] + p.lnoutb[i];
    __syncthreads();
    const int HT = (RV + 15) / 16;  // 3143 tiles, full K per wave
    for (int u = wid; u < HT; u += NWAVES)
      gemv16(p.head, RE, RV, sh, u * 16, 0, RE, p.logits);
  }
}

extern "C" void kernel_launch(void *const *d_in, const int *in_sizes, int n_in,
                              void *d_out, int out_size, void *d_ws,
                              size_t ws_size, hipStream_t stream) {
  (void)in_sizes; (void)n_in; (void)out_size; (void)ws_size;
  Params P;
  P.ctx    = (const float *)d_in[0];
  P.state  = (const float *)d_in[1];
  P.ln0w   = (const float *)d_in[2];
  P.ln0b   = (const float *)d_in[3];
  P.ln1w   = (const float *)d_in[4];
  P.ln1b   = (const float *)d_in[5];
  P.ln2w   = (const float *)d_in[6];
  P.ln2b   = (const float *)d_in[7];
  P.td     = (const float *)d_in[8];
  P.tf     = (const float *)d_in[9];
  P.tmk    = (const float *)d_in[10];
  P.tmv    = (const float *)d_in[11];
  P.tmr    = (const float *)d_in[12];
  P.kw     = (const float *)d_in[13];
  P.vw     = (const float *)d_in[14];
  P.rw     = (const float *)d_in[15];
  P.ow     = (const float *)d_in[16];
  P.ftmk   = (const float *)d_in[17];
  P.ftmr   = (const float *)d_in[18];
  P.fkw    = (const float *)d_in[19];
  P.frw    = (const float *)d_in[20];
  P.fvw    = (const float *)d_in[21];
  P.lnoutw = (const float *)d_in[22];
  P.lnoutb = (const float *)d_in[23];
  P.head   = (const float *)d_in[24];
  P.logits = (float *)d_out;
  P.ostate = (float *)d_out + RV;
  P.cnt = (unsigned *)d_ws;
  float *f = (float *)((char *)d_ws + 256);
  P.x0 = f;  f += RE;
  P.x1 = f;  f += RE;
  P.pk = f;  f += P2 * RE;
  P.pv = f;  f += P2 * RE;
  P.pr = f;  f += P2 * RE;
  P.po = f;  f += P4 * RE;
  P.pfr = f; f += P5R * RE;
  P.pkk = f; f += P5K * RH;
  P.pfv = f; f += P6 * RE;
  hipMemsetAsync(d_ws, 0, 256, stream);  // reset grid-sync counter (capture-safe)
  rwkv_fused<<<dim3(NWG), dim3(TPB), 0, stream>>>(P);
}